// TemporalPropagator_79328045957539
// MI455X (gfx1250) — compile-verified
//
#include <hip/hip_runtime.h>
#include <math.h>

// ---- problem constants (match reference) ----
#define BATCH 8
#define TLEN  4096
#define DIM   256
#define TWOD  512
#define BT    (BATCH * TLEN)     // 32768
#define CHUNK 128
#define NCH   (TLEN / CHUNK)     // 32
#define EPSF  1e-7f

typedef __attribute__((ext_vector_type(2))) float v2f;
typedef __attribute__((ext_vector_type(8))) float v8f;
typedef __attribute__((__vector_size__(4 * sizeof(int)))) int v4i;

#define GAS __attribute__((address_space(1)))   // global
#define LAS __attribute__((address_space(3)))   // LDS

#if defined(__has_builtin)
#if __has_builtin(__builtin_amdgcn_global_load_async_to_lds_b128)
#define HAVE_ASYNC_LDS 1
#endif
#endif

// ---------------- helpers ----------------
__device__ __forceinline__ float softplusf(float x) {
    return fmaxf(x, 0.0f) + log1pf(expf(-fabsf(x)));
}

__device__ __forceinline__ v8f wmma4(v2f a, v2f b, v8f c) {
    // D = A(16x4,f32) * B(4x16,f32) + C(16x16,f32)   [V_WMMA_F32_16X16X4_F32]
    return __builtin_amdgcn_wmma_f32_16x16x4_f32(
        /*neg_a=*/false, a, /*neg_b=*/false, b,
        /*c_mod=*/(short)0, c, /*reuse_a=*/false, /*reuse_b=*/false);
}

// complex decay/forcing from z = (lre + i*lim) * dt
__device__ __forceinline__ void decay_forcing(float lre, float lim, float dtv,
                                              float& dre, float& dim_,
                                              float& fre, float& fim) {
    float zr = lre * dtv, zi = lim * dtv;
    float er = expf(zr);
    float sn, cs;
    __sincosf(zi, &sn, &cs);
    dre = er * cs;
    dim_ = er * sn;
    float srcr = zr + EPSF;
    float sg = (srcr > 0.0f) ? 1.0f : ((srcr < 0.0f) ? -1.0f : 0.0f);
    float denr = zr + EPSF * sg;
    float deni = zi;
    float nr = dre - 1.0f, ni = dim_;
    float inv = 1.0f / (denr * denr + deni * deni);
    fre = (nr * denr + ni * deni) * inv;
    fim = (ni * denr - nr * deni) * inv;
}

// ---------------- K1: x_in GEMM + forcing -> X_scan (split re/im planes) ----------------
// grid: BT/16 blocks, 256 threads (8 waves). Wave w handles 64 cols (4 WMMA tiles).
__global__ void k1_gemm_in(const float* __restrict__ x,
                           const float* __restrict__ dt,
                           const float* __restrict__ decay_re,
                           const float* __restrict__ decay_im,
                           const float* __restrict__ W_in,
                           const float* __restrict__ b_in,
                           float* __restrict__ Xre,
                           float* __restrict__ Xim) {
    __shared__ float smem[16 * TWOD]; // phase A: A-tile 16x256; phase B: O-tile 16x512
    const int m0 = blockIdx.x * 16;
    const int tid = threadIdx.x;

    // stage A (x rows, contiguous 16KB) into LDS
#ifdef HAVE_ASYNC_LDS
    {
        const float* gsrc = x + (size_t)m0 * DIM;
#pragma unroll
        for (int i = 0; i < 4; ++i) {
            int f4 = (tid + i * 256) * 4; // float offset, 16B chunks
            __builtin_amdgcn_global_load_async_to_lds_b128(
                (GAS v4i*)(gsrc + f4),
                (LAS v4i*)((LAS float*)smem + f4), 0, 0);
        }
        asm volatile("s_wait_asynccnt 0x0" ::: "memory");
    }
#else
    for (int e = tid; e < 16 * DIM; e += 256)
        smem[e] = x[(size_t)m0 * DIM + e];
#endif
    __syncthreads();

    const int wave = tid >> 5, lane = tid & 31;
    const int lrow = lane & 15, khalf = lane >> 4;
    const int colBase = wave * 64;

    // warm L2 for this wave's weight rows
    __builtin_prefetch(W_in + (size_t)(colBase + lrow) * TWOD, 0, 1);

    v8f acc[4] = {};
    for (int k0 = 0; k0 < DIM; k0 += 4) {
        v2f a;
        a.x = smem[lrow * DIM + k0 + 2 * khalf];
        a.y = smem[lrow * DIM + k0 + 2 * khalf + 1];
#pragma unroll
        for (int t = 0; t < 4; ++t) {
            int col = colBase + t * 16 + lrow;
            const float* wp = W_in + (size_t)col * TWOD + k0 + 2 * khalf;
            v2f b; b.x = wp[0]; b.y = wp[1];
            acc[t] = wmma4(a, b, acc[t]);
        }
    }
    __syncthreads(); // done reading A region; alias smem as O-tile

#pragma unroll
    for (int t = 0; t < 4; ++t)
#pragma unroll
        for (int i = 0; i < 8; ++i)
            smem[(i + 8 * khalf) * TWOD + colBase + t * 16 + lrow] = acc[t][i];
    __syncthreads();

    // elementwise: thread d handles one channel across all 16 rows
    const int d = tid;
    const float lre = -softplusf(decay_re[d]);
    const float lim = decay_im[d];
    const float bre = b_in[d], bim = b_in[d + DIM];
    for (int r = 0; r < 16; ++r) {
        int m = m0 + r;
        float dre, dim_, fre, fim;
        decay_forcing(lre, lim, dt[m], dre, dim_, fre, fim);
        float xr = smem[r * TWOD + d] + bre;
        float xi = smem[r * TWOD + DIM + d] + bim;
        size_t idx = (size_t)m * DIM + d;
        Xre[idx] = xr * fre - xi * fim;
        Xim[idx] = xr * fim + xi * fre;
    }
}

// ---------------- K2a: per-chunk aggregates (a = prod decay, b = local scan end) ----------------
// grid: BATCH*NCH blocks, 256 threads (thread = channel d)
__global__ void k2a_aggregate(const float* __restrict__ Xre,
                              const float* __restrict__ Xim,
                              const float* __restrict__ dt,
                              const float* __restrict__ decay_re,
                              const float* __restrict__ decay_im,
                              float* __restrict__ carry) {
    const int blk = blockIdx.x;
    const int b = blk / NCH, ch = blk % NCH;
    const int d = threadIdx.x;
    const float lre = -softplusf(decay_re[d]);
    const float lim = decay_im[d];
    float ar = 1.0f, ai = 0.0f, br = 0.0f, bi = 0.0f;
    const int base = b * TLEN + ch * CHUNK;
    for (int s = 0; s < CHUNK; ++s) {
        int m = base + s;
        float dtv = dt[m];
        float zr = lre * dtv, zi = lim * dtv;
        float er = expf(zr);
        float sn, cs; __sincosf(zi, &sn, &cs);
        float dre = er * cs, dim_ = er * sn;
        size_t idx = (size_t)m * DIM + d;
        float Xr = Xre[idx], Xi = Xim[idx];
        float nbr = br * dre - bi * dim_ + Xr;
        float nbi = br * dim_ + bi * dre + Xi;
        br = nbr; bi = nbi;
        float nar = ar * dre - ai * dim_;
        float nai = ar * dim_ + ai * dre;
        ar = nar; ai = nai;
    }
    float4 cv; cv.x = ar; cv.y = ai; cv.z = br; cv.w = bi;
    ((float4*)carry)[(size_t)blk * DIM + d] = cv;
}

// ---------------- K2b: exclusive prefix over chunk aggregates ----------------
// grid: BATCH blocks, 256 threads
__global__ void k2b_prefix(const float* __restrict__ carry,
                           float* __restrict__ pref) {
    const int b = blockIdx.x;
    const int d = threadIdx.x;
    float pbr = 0.0f, pbi = 0.0f;
    for (int ch = 0; ch < NCH; ++ch) {
        size_t idx = (size_t)(b * NCH + ch) * DIM + d;
        float4 agg = ((const float4*)carry)[idx];
        float2 p; p.x = pbr; p.y = pbi;
        ((float2*)pref)[idx] = p; // exclusive prefix (flux value just before chunk)
        float nbr = pbr * agg.x - pbi * agg.y + agg.z;
        float nbi = pbr * agg.y + pbi * agg.x + agg.w;
        pbr = nbr; pbi = nbi;
    }
}

// ---------------- K2c: apply prefix, write flux in place over X planes ----------------
__global__ void k2c_apply(float* __restrict__ Xre,
                          float* __restrict__ Xim,
                          const float* __restrict__ dt,
                          const float* __restrict__ decay_re,
                          const float* __restrict__ decay_im,
                          const float* __restrict__ pref) {
    const int blk = blockIdx.x;
    const int b = blk / NCH, ch = blk % NCH;
    const int d = threadIdx.x;
    const float lre = -softplusf(decay_re[d]);
    const float lim = decay_im[d];
    float2 p = ((const float2*)pref)[(size_t)blk * DIM + d];
    float br = p.x, bi = p.y;
    const int base = b * TLEN + ch * CHUNK;
    for (int s = 0; s < CHUNK; ++s) {
        int m = base + s;
        float dtv = dt[m];
        float zr = lre * dtv, zi = lim * dtv;
        float er = expf(zr);
        float sn, cs; __sincosf(zi, &sn, &cs);
        float dre = er * cs, dim_ = er * sn;
        size_t idx = (size_t)m * DIM + d;
        float Xr = Xre[idx], Xi = Xim[idx];
        float nbr = br * dre - bi * dim_ + Xr;
        float nbi = br * dim_ + bi * dre + Xi;
        br = nbr; bi = nbi;
        Xre[idx] = br;
        Xim[idx] = bi;
    }
}

// ---------------- K3: output GEMMs ([W_out; W_gate], 768 cols) + elementwise tail ----------------
// grid: BT/16 blocks, 256 threads (8 waves; wave w -> 96 cols = 6 tiles)
__global__ void k3_gemm_out(const float* __restrict__ Fre,
                            const float* __restrict__ Fim,
                            const float* __restrict__ dt,
                            const float* __restrict__ W_out,
                            const float* __restrict__ b_out,
                            const float* __restrict__ W_gate,
                            const float* __restrict__ b_gate,
                            const float* __restrict__ lam_re,
                            const float* __restrict__ lam_im,
                            float* __restrict__ out) {
    __shared__ float smem[16 * 768]; // phase A: s_cat tile 16x512; phase B: O-tile 16x768
    const int m0 = blockIdx.x * 16;
    const int tid = threadIdx.x;

    // stage A: s_cat row-major [re | im]; both planes are row-contiguous (1KB chunks)
#ifdef HAVE_ASYNC_LDS
    {
#pragma unroll
        for (int i = 0; i < 4; ++i) {
            int f4 = tid + i * 256;     // 0..1023 float4 chunks per plane
            int row = f4 >> 6;          // 64 chunks per 256-float row
            int cw = (f4 & 63) * 4;     // float offset in row
            __builtin_amdgcn_global_load_async_to_lds_b128(
                (GAS v4i*)(Fre + (size_t)(m0 + row) * DIM + cw),
                (LAS v4i*)((LAS float*)smem + row * TWOD + cw), 0, 0);
            __builtin_amdgcn_global_load_async_to_lds_b128(
                (GAS v4i*)(Fim + (size_t)(m0 + row) * DIM + cw),
                (LAS v4i*)((LAS float*)smem + row * TWOD + DIM + cw), 0, 0);
        }
        asm volatile("s_wait_asynccnt 0x0" ::: "memory");
    }
#else
    for (int e = tid; e < 16 * DIM; e += 256) {
        int r = e >> 8, d = e & 255;
        smem[r * TWOD + d]       = Fre[(size_t)(m0 + r) * DIM + d];
        smem[r * TWOD + DIM + d] = Fim[(size_t)(m0 + r) * DIM + d];
    }
#endif
    __syncthreads();

    const int wave = tid >> 5, lane = tid & 31;
    const int lrow = lane & 15, khalf = lane >> 4;
    const int colBase = wave * 96;

    {   // warm L2 for this wave's weight rows
        int col = colBase + lrow;
        const float* wp = (col < TWOD) ? (W_out + (size_t)col * TWOD)
                                       : (W_gate + (size_t)(col - TWOD) * TWOD);
        __builtin_prefetch(wp, 0, 1);
    }

    v8f acc[6] = {};
    for (int k0 = 0; k0 < TWOD; k0 += 4) {
        v2f a;
        a.x = smem[lrow * TWOD + k0 + 2 * khalf];
        a.y = smem[lrow * TWOD + k0 + 2 * khalf + 1];
#pragma unroll
        for (int t = 0; t < 6; ++t) {
            int col = colBase + t * 16 + lrow; // 0..767
            const float* wp = (col < TWOD)
                ? (W_out + (size_t)col * TWOD)
                : (W_gate + (size_t)(col - TWOD) * TWOD);
            v2f b; b.x = wp[k0 + 2 * khalf]; b.y = wp[k0 + 2 * khalf + 1];
            acc[t] = wmma4(a, b, acc[t]);
        }
    }
    __syncthreads(); // done reading A region; alias smem as O-tile (16x768)

#pragma unroll
    for (int t = 0; t < 6; ++t)
#pragma unroll
        for (int i = 0; i < 8; ++i)
            smem[(i + 8 * khalf) * 768 + colBase + t * 16 + lrow] = acc[t][i];
    __syncthreads();

    // elementwise tail: thread d over 16 rows
    const int d = tid;
    const float lpre = -softplusf(lam_re[d]);
    const float lpim = lam_im[d];
    const float bor = b_out[d], boi = b_out[d + DIM], bg = b_gate[d];
    for (int r = 0; r < 16; ++r) {
        int m = m0 + r;
        float dpr, dpi, fpr, fpi;
        decay_forcing(lpre, lpim, dt[m], dpr, dpi, fpr, fpi);
        float sre = smem[r * 768 + d] + bor;
        float sim = smem[r * 768 + DIM + d] + boi;
        float gl = smem[r * 768 + TWOD + d] + bg;
        float g = 1.0f / (1.0f + expf(-gl));
        g = g * 0.98f + 0.01f;
        float* o = out + (size_t)m * (7 * DIM);
        o[d]            = sre;
        o[DIM + d]      = sim;
        o[2 * DIM + d]  = g;
        o[3 * DIM + d]  = dpr;
        o[4 * DIM + d]  = dpi;
        o[5 * DIM + d]  = fpr;
        o[6 * DIM + d]  = fpi;
    }
}

// ---------------- launcher ----------------
extern "C" void kernel_launch(void* const* d_in, const int* in_sizes, int n_in,
                              void* d_out, int out_size, void* d_ws, size_t ws_size,
                              hipStream_t stream) {
    const float* x        = (const float*)d_in[0];
    const float* dt       = (const float*)d_in[1];
    const float* decay_re = (const float*)d_in[2];
    const float* decay_im = (const float*)d_in[3];
    const float* W_in     = (const float*)d_in[4];
    const float* b_in     = (const float*)d_in[5];
    const float* W_out    = (const float*)d_in[6];
    const float* b_out    = (const float*)d_in[7];
    const float* W_gate   = (const float*)d_in[8];
    const float* b_gate   = (const float*)d_in[9];
    const float* lam_re   = (const float*)d_in[10];
    const float* lam_im   = (const float*)d_in[11];
    float* out = (float*)d_out;

    // workspace layout (floats): Xre [BT*D] | Xim [BT*D] | carry [B*NCH*D*4] | pref [B*NCH*D*2]
    float* Xre   = (float*)d_ws;
    float* Xim   = Xre + (size_t)BT * DIM;
    float* carry = Xim + (size_t)BT * DIM;
    float* pref  = carry + (size_t)BATCH * NCH * DIM * 4;

    k1_gemm_in<<<BT / 16, 256, 0, stream>>>(x, dt, decay_re, decay_im, W_in, b_in, Xre, Xim);
    k2a_aggregate<<<BATCH * NCH, 256, 0, stream>>>(Xre, Xim, dt, decay_re, decay_im, carry);
    k2b_prefix<<<BATCH, 256, 0, stream>>>(carry, pref);
    k2c_apply<<<BATCH * NCH, 256, 0, stream>>>(Xre, Xim, dt, decay_re, decay_im, pref);
    k3_gemm_out<<<BT / 16, 256, 0, stream>>>(Xre, Xim, dt, W_out, b_out, W_gate, b_gate,
                                             lam_re, lam_im, out);
}